// L_GCN_79164837200177
// MI455X (gfx1250) — compile-verified
//
#include <hip/hip_runtime.h>
#include <cstdint>

// Problem constants
#define Bq   4
#define Sq   256
#define Dq   768
#define DEPq 100
#define Hq   12
#define DKq  64

typedef __attribute__((ext_vector_type(2))) float v2f;
typedef __attribute__((ext_vector_type(8))) float v8f;
typedef __attribute__((ext_vector_type(4))) unsigned int u32x4;
typedef __attribute__((ext_vector_type(8))) unsigned int u32x8;

// ---------------------------------------------------------------------------
// WMMA helper: D = A(16x4 f32) * B(4x16 f32) + C (16x16 f32)
// ---------------------------------------------------------------------------
static __device__ __forceinline__ v8f wmma4(v2f a, v2f b, v8f c) {
  return __builtin_amdgcn_wmma_f32_16x16x4_f32(
      /*neg_a=*/false, a, /*neg_b=*/false, b,
      /*c_mod=*/(short)0, c, /*reuse_a=*/false, /*reuse_b=*/false);
}

// ---------------------------------------------------------------------------
// TDM: 2D tile load Global -> LDS via tensor_load_to_lds (D# packed by hand).
// tile_d0 = elements per row (4-byte elems), tile_d1 = rows,
// row_stride = tensor dim0 stride in elements. All args wave-uniform.
// ---------------------------------------------------------------------------
static __device__ __forceinline__ void tdm_load_2d(unsigned lds_byte_off,
                                                   const void* gaddr,
                                                   unsigned tile_d0,
                                                   unsigned tile_d1,
                                                   unsigned row_stride) {
  unsigned long long ga = (unsigned long long)gaddr;
  u32x4 g0;
  g0.x = 1u;                                   // count=1, user descriptor
  g0.y = lds_byte_off;                         // lds_addr
  g0.z = (unsigned)(ga & 0xFFFFFFFFu);         // global_addr[31:0]
  g0.w = (unsigned)((ga >> 32) & 0x1FFFFFFu)   // global_addr[56:32]
         | 0x80000000u;                        // type=2 (bits 127:126)
  u32x8 g1;
  g1.s0 = 2u << 16;                            // data_size=2 (4 bytes)
  g1.s1 = (tile_d0 & 0xFFFFu) << 16;           // tensor_dim0[15:0] @ bits63:48
  g1.s2 = (tile_d1 & 0xFFFFu) << 16;           // tensor_dim0 hi=0; tensor_dim1 lo
  g1.s3 = (tile_d0 & 0xFFFFu) << 16;           // tensor_dim1 hi=0; tile_dim0
  g1.s4 = (tile_d1 & 0xFFFFu);                 // tile_dim1; tile_dim2=0
  g1.s5 = row_stride;                          // tensor_dim0_stride[31:0]
  g1.s6 = 0u;                                  // stride0 hi; dim1_stride lo
  g1.s7 = 0u;
  asm volatile("tensor_load_to_lds %0, %1" :: "s"(g0), "s"(g1) : "memory");
}

static __device__ __forceinline__ unsigned lds_off(const void* p) {
  return (unsigned)(unsigned long long)p;      // low 32 bits = LDS byte offset
}

// ---------------------------------------------------------------------------
// Kernel 1: fold w_rel into W_e:  U_t[h][k] = sum_d W_e[k, h*64+d] * w_rel[d]
// Stored transposed [16][100] (h-major, zero-padded heads 12..15).
// ---------------------------------------------------------------------------
__global__ __launch_bounds__(256) void fold_u_kernel(
    const float* __restrict__ W_e, const float* __restrict__ w_rel,
    float* __restrict__ U_t) {
  int idx = blockIdx.x * blockDim.x + threadIdx.x;   // 16 * 100
  if (idx >= 16 * DEPq) return;
  int n = idx / DEPq, k = idx % DEPq;
  float acc = 0.0f;
  if (n < Hq) {
    #pragma unroll 4
    for (int d = 0; d < DKq; ++d) acc += W_e[k * Dq + n * DKq + d] * w_rel[d];
  }
  U_t[n * DEPq + k] = acc;
}

// ---------------------------------------------------------------------------
// Kernel 2: V = token @ W_v   ([1024 x 768] @ [768 x 768], f32 WMMA)
// One 16x16 tile per wave; 8 waves/block; tiles = 64 x 48.
// ---------------------------------------------------------------------------
__global__ __launch_bounds__(256) void vproj_kernel(
    const float* __restrict__ tok, const float* __restrict__ Wv,
    float* __restrict__ V) {
  const int wave = blockIdx.x * (blockDim.x >> 5) + (threadIdx.x >> 5);
  const int lane = threadIdx.x & 31;
  const int kg = lane >> 4, lm = lane & 15;
  const int mt = wave / (Dq / 16);            // 0..63
  const int nt = wave % (Dq / 16);            // 0..47
  const int row = mt * 16 + lm;
  const int col = nt * 16 + lm;
  v8f acc = {};
  for (int kb = 0; kb < Dq; kb += 4) {
    const int ka = kb + 2 * kg;
    v2f a = *(const v2f*)(tok + (size_t)row * Dq + ka);
    v2f b;
    b.x = Wv[(size_t)(ka + 0) * Dq + col];
    b.y = Wv[(size_t)(ka + 1) * Dq + col];
    acc = wmma4(a, b, acc);
  }
  const int mrow = mt * 16 + kg * 8;          // lanes 0-15: M=r, 16-31: M=r+8
  #pragma unroll
  for (int r = 0; r < 8; ++r)
    V[(size_t)(mrow + r) * Dq + col] = acc[r];
}

// ---------------------------------------------------------------------------
// Block-wide reductions (256 threads, 8 waves)
// ---------------------------------------------------------------------------
static __device__ __forceinline__ float block_max(float v, float* red, int tid) {
  #pragma unroll
  for (int o = 16; o > 0; o >>= 1) v = fmaxf(v, __shfl_xor(v, o, 32));
  if ((tid & 31) == 0) red[tid >> 5] = v;
  __syncthreads();
  float r = red[0];
  #pragma unroll
  for (int w = 1; w < 8; ++w) r = fmaxf(r, red[w]);
  __syncthreads();
  return r;
}

static __device__ __forceinline__ float block_sum(float v, float* red, int tid) {
  #pragma unroll
  for (int o = 16; o > 0; o >>= 1) v += __shfl_xor(v, o, 32);
  if ((tid & 31) == 0) red[tid >> 5] = v;
  __syncthreads();
  float r = red[0];
  #pragma unroll
  for (int w = 1; w < 8; ++w) r += red[w];
  __syncthreads();
  return r;
}

// ---------------------------------------------------------------------------
// Kernel 3: per (b,i): scores = edge[b,i,:,:] @ U_t^T (WMMA, TDM-staged edge),
// relu + mask + softmax over j, write probs[b,h,i,j].
// Block = (b*256+i), 256 threads, wave w owns j-tiles {2w, 2w+1}.
// ---------------------------------------------------------------------------
__global__ __launch_bounds__(256) void scores_softmax_kernel(
    const float* __restrict__ edge, const int* __restrict__ mask,
    const float* __restrict__ U_t, float* __restrict__ probs) {
  __shared__ float sEdge[8 * 2 * 16 * DEPq];   // 102400 B, per-wave regions
  __shared__ float sU[16 * DEPq];              // 6400 B
  __shared__ float sScore[Hq * Sq];            // 12288 B
  __shared__ float sRed[8];

  const int bi = blockIdx.x;                   // b*256 + i
  const int tid = threadIdx.x;
  const int wv = tid >> 5, lane = tid & 31;
  const int kg = lane >> 4, lm = lane & 15;
  const int jt0 = wv * 2;

  // stage U (B-matrix, shared by all waves)
  for (int t = tid; t < 16 * DEPq; t += 256) sU[t] = U_t[t];

  // issue both TDM tile loads for this wave (16 rows x 100 f32 each)
  const unsigned eBase = lds_off(sEdge);
  #pragma unroll
  for (int q = 0; q < 2; ++q) {
    const float* g = edge + ((size_t)bi * Sq + (size_t)(jt0 + q) * 16) * DEPq;
    tdm_load_2d(eBase + (unsigned)(wv * 2 + q) * 16 * DEPq * 4, g,
                DEPq, 16, DEPq);
  }
  __syncthreads();                              // sU visible to all waves

  #pragma unroll
  for (int q = 0; q < 2; ++q) {
    if (q == 0) __builtin_amdgcn_s_wait_tensorcnt(1);   // tile0 landed
    else        __builtin_amdgcn_s_wait_tensorcnt(0);   // tile1 landed
    const float* eb = &sEdge[(wv * 2 + q) * 16 * DEPq];
    v8f c = {};
    for (int kb = 0; kb < DEPq; kb += 4) {
      const int ka = kb + 2 * kg;
      v2f a = *(const v2f*)(eb + lm * DEPq + ka);    // A: edge rows (j, k)
      v2f b = *(const v2f*)(sU + lm * DEPq + ka);    // B: U_t[h][k]
      c = wmma4(a, b, c);
    }
    if (lm < Hq) {
      #pragma unroll
      for (int r = 0; r < 8; ++r) {
        const int j = (jt0 + q) * 16 + r + kg * 8;
        sScore[lm * Sq + j] = c[r];
      }
    }
  }
  __syncthreads();

  // fused mask + relu + softmax over j (tid == j), then write probs
  const int j = tid;
  const int b = bi >> 8, i = bi & 255;
  const int mk = mask[(size_t)bi * Sq + j];
  for (int h = 0; h < Hq; ++h) {
    float s = sScore[h * Sq + j];
    s = (mk == 0) ? -1.0e6f : fmaxf(s, 0.0f);
    const float m = block_max(s, sRed, tid);
    const float e = __expf(s - m);
    const float sum = block_sum(e, sRed, tid);
    probs[(((size_t)b * Hq + h) * Sq + i) * Sq + j] = e / sum;
  }
}

// ---------------------------------------------------------------------------
// Kernel 4: ctx = probs[b,h] @ V[b,:,h,:]  (256x256 @ 256x64, WMMA),
// out = relu(token + ctx). Block = (b*12+h); V slice TDM-staged into LDS.
// Wave w: i-tiles {2w,2w+1} x 4 n-tiles.
// ---------------------------------------------------------------------------
__global__ __launch_bounds__(256) void attn_ctx_kernel(
    const float* __restrict__ probs, const float* __restrict__ V,
    const float* __restrict__ tok, float* __restrict__ out) {
  __shared__ float sV[Sq * DKq];               // 64 KB
  const int bh = blockIdx.x;
  const int b = bh / Hq, h = bh % Hq;
  const int tid = threadIdx.x;
  const int wv = tid >> 5, lane = tid & 31;
  const int kg = lane >> 4, lm = lane & 15;

  if (wv == 0) {                               // one TDM for the whole block
    const float* g = V + (size_t)b * Sq * Dq + (size_t)h * DKq;
    tdm_load_2d(lds_off(sV), g, DKq, Sq, Dq);  // 256 rows x 64 f32, stride 768
    __builtin_amdgcn_s_wait_tensorcnt(0);
  }
  __syncthreads();

  v8f acc[2][4];
  v8f zero = {};
  #pragma unroll
  for (int it = 0; it < 2; ++it)
    #pragma unroll
    for (int nt = 0; nt < 4; ++nt) acc[it][nt] = zero;

  const float* p0 = probs + ((size_t)bh * Sq + (size_t)(wv * 2 + 0) * 16 + lm) * Sq;
  const float* p1 = probs + ((size_t)bh * Sq + (size_t)(wv * 2 + 1) * 16 + lm) * Sq;

  for (int kb = 0; kb < Sq; kb += 4) {
    const int ka = kb + 2 * kg;
    v2f a0 = *(const v2f*)(p0 + ka);
    v2f a1 = *(const v2f*)(p1 + ka);
    #pragma unroll
    for (int nt = 0; nt < 4; ++nt) {
      v2f bf;
      bf.x = sV[(ka + 0) * DKq + nt * 16 + lm];
      bf.y = sV[(ka + 1) * DKq + nt * 16 + lm];
      acc[0][nt] = wmma4(a0, bf, acc[0][nt]);
      acc[1][nt] = wmma4(a1, bf, acc[1][nt]);
    }
  }

  // epilogue: out = relu(token + ctx); C rows are contiguous 16-float segments
  #pragma unroll
  for (int it = 0; it < 2; ++it)
    #pragma unroll
    for (int nt = 0; nt < 4; ++nt)
      #pragma unroll
      for (int r = 0; r < 8; ++r) {
        const int i = (wv * 2 + it) * 16 + r + kg * 8;
        const int d = h * DKq + nt * 16 + lm;
        const size_t idx = ((size_t)b * Sq + i) * Dq + d;
        out[idx] = fmaxf(tok[idx] + acc[it][nt][r], 0.0f);
      }
}

// ---------------------------------------------------------------------------
// Host-side launcher
// ---------------------------------------------------------------------------
extern "C" void kernel_launch(void* const* d_in, const int* in_sizes, int n_in,
                              void* d_out, int out_size, void* d_ws, size_t ws_size,
                              hipStream_t stream) {
  (void)in_sizes; (void)n_in; (void)out_size; (void)ws_size;
  const float* tok  = (const float*)d_in[0];   // [4,256,768]
  const float* edge = (const float*)d_in[1];   // [4,256,256,100]
  const int*   msk  = (const int*)  d_in[2];   // [4,256,256]
  const float* Wv   = (const float*)d_in[3];   // [768,768]
  const float* We   = (const float*)d_in[4];   // [100,768]
  const float* wrel = (const float*)d_in[5];   // [64]
  float* out = (float*)d_out;                  // [4,256,768]

  float* U_t   = (float*)d_ws;                 // 16*100
  float* Vws   = U_t + 16 * DEPq;              // 1024*768
  float* probs = Vws + (size_t)Bq * Sq * Dq;   // 4*12*256*256

  fold_u_kernel<<<(16 * DEPq + 255) / 256, 256, 0, stream>>>(We, wrel, U_t);
  vproj_kernel<<<(64 * 48) / 8, 256, 0, stream>>>(tok, Wv, Vws);
  scores_softmax_kernel<<<Bq * Sq, 256, 0, stream>>>(edge, msk, U_t, probs);
  attn_ctx_kernel<<<Bq * Hq, 256, 0, stream>>>(probs, Vws, tok, out);
}